// HWnet_base_9096740733131
// MI455X (gfx1250) — compile-verified
//
#include <hip/hip_runtime.h>
#include <math.h>

// Problem constants (match reference: B=65536, T=4096, D=128, EDGE=8)
#define T_TAB   4096
#define D_DIM   128
#define EDGE    8
#define WIN     (2 * EDGE + 1)          // 17
#define WAVESZ  32                      // gfx1250 is wave32-only
#define WPB     8                       // waves (samples) per block
#define BLOCK   (WPB * WAVESZ)          // 256 threads

typedef float vf4 __attribute__((ext_vector_type(4)));
typedef int   i4v __attribute__((vector_size(16)));   // matches builtin param

// ---- CDNA5 async global->LDS staging (ASYNCcnt path) -----------------------
#if defined(__has_builtin)
#  if __has_builtin(__builtin_amdgcn_global_load_async_to_lds_b128)
#    define USE_ASYNC_LDS 1
#  endif
#  if __has_builtin(__builtin_amdgcn_s_wait_asynccnt)
#    define HAVE_WAIT_ASYNC 1
#  endif
#endif

typedef __attribute__((address_space(1))) i4v GV4;   // global int4
typedef __attribute__((address_space(3))) i4v LV4;   // LDS int4

__global__ __launch_bounds__(BLOCK) void hwnet_kernel(
    const float* __restrict__ x_in,        // [B]
    const float* __restrict__ eval_tab,    // [T] sorted ascending
    const float* __restrict__ take_tab,    // [T]
    const float* __restrict__ vec_tab,     // [T, D]
    float* __restrict__ out,               // [B, D]
    int nB)
{
    __shared__ __align__(16) float s_eval[T_TAB];      // 16 KB

    // ---- stage sorted evaluate_table into LDS -----------------------------
    {
        const vf4* g = (const vf4*)eval_tab;
        vf4*       s = (vf4*)s_eval;
#ifdef USE_ASYNC_LDS
        // 1024 x b128 transfers, 4 per thread; tracked with ASYNCcnt
        for (int i = threadIdx.x; i < T_TAB / 4; i += BLOCK) {
            __builtin_amdgcn_global_load_async_to_lds_b128(
                (GV4*)(g + i), (LV4*)(s + i), /*offset=*/0, /*cpol=*/0);
        }
#  ifdef HAVE_WAIT_ASYNC
        __builtin_amdgcn_s_wait_asynccnt(0);
#  else
        asm volatile("s_wait_asynccnt 0" ::: "memory");
#  endif
#else
        for (int i = threadIdx.x; i < T_TAB / 4; i += BLOCK) s[i] = g[i];
#endif
    }
    __syncthreads();

    const int wave   = threadIdx.x >> 5;
    const int lane   = threadIdx.x & (WAVESZ - 1);
    const int sample = blockIdx.x * WPB + wave;
    if (sample >= nB) return;

    const float x = x_in[sample];          // wave-uniform -> scalarized load

    // ---- branchless binary search in LDS: largest pos with eval[pos] <= x -
    int pos = -1;
    #pragma unroll
    for (int step = T_TAB; step >= 1; step >>= 1) {
        int np = pos + step;
        if (np < T_TAB && s_eval[np] <= x) pos = np;
    }
    // nearest index (argmin over squared distance; ties -> smaller index)
    int idx;
    if (pos < 0)                 idx = 0;
    else if (pos >= T_TAB - 1)   idx = T_TAB - 1;
    else {
        float d0 = x - s_eval[pos];        // >= 0
        float d1 = s_eval[pos + 1] - x;    // >  0
        idx = (d1 < d0) ? pos + 1 : pos;
    }

    const float tk = take_tab[idx];        // unclipped idx (matches reference)

    int base = idx;
    base = base < EDGE ? EDGE : base;
    base = base > (T_TAB - EDGE - 1) ? (T_TAB - EDGE - 1) : base;
    base -= EDGE;                          // window start, in [0, T-17]

    // ---- lane-parallel softmax over the 17-tap window ---------------------
    float v = -INFINITY;
    if (lane < WIN) {
        float d = x - s_eval[base + lane];
        v = -(d * d) * tk;
    }
    float m = v;
    #pragma unroll
    for (int o = WAVESZ / 2; o >= 1; o >>= 1)
        m = fmaxf(m, __shfl_xor(m, o, WAVESZ));
    float e = __expf(v - m);               // lanes >= 17: exp(-inf) = 0
    float ssum = e;
    #pragma unroll
    for (int o = WAVESZ / 2; o >= 1; o >>= 1)
        ssum += __shfl_xor(ssum, o, WAVESZ);
    const float inv = 1.0f / ssum;

    // ---- weighted gather-reduction: each lane owns 4 output columns -------
    const vf4* __restrict__ vt4 = (const vf4*)vec_tab;     // row = 32 vf4
    vf4 acc = (vf4)(0.0f);
    #pragma unroll
    for (int w = 0; w < WIN; ++w) {
        float s = __shfl(e, w, WAVESZ);    // constant lane -> readlane bcast
        vf4 row = vt4[(base + w) * (D_DIM / 4) + lane];    // global_load_b128
        acc += row * s;                    // v_fmac_f32 x4
    }
    acc *= inv;

    // stream result (written once; don't pollute L2 holding vector_table)
    __builtin_nontemporal_store(acc, &((vf4*)out)[sample * (D_DIM / 4) + lane]);
}

extern "C" void kernel_launch(void* const* d_in, const int* in_sizes, int n_in,
                              void* d_out, int out_size, void* d_ws, size_t ws_size,
                              hipStream_t stream) {
    const float* x  = (const float*)d_in[0];   // [B]
    const float* ev = (const float*)d_in[1];   // [T] (T == 4096, sorted)
    const float* tk = (const float*)d_in[2];   // [T]
    const float* vt = (const float*)d_in[3];   // [T, D]
    // d_in[4] = edge_size (== 8, compile-time constant here)

    const int B    = in_sizes[0];
    const int grid = (B + WPB - 1) / WPB;

    hipLaunchKernelGGL(hwnet_kernel, dim3(grid), dim3(BLOCK), 0, stream,
                       x, ev, tk, vt, (float*)d_out, B);
}